// GaussianDropoutClass_68109591380478
// MI455X (gfx1250) — compile-verified
//
#include <hip/hip_runtime.h>
#include <hip/hip_bf16.h>
#include <stdint.h>
#include <math.h>

namespace {

struct U2 { unsigned a, b; };

__host__ __device__ constexpr unsigned rotl32(unsigned x, int r) {
  return (x << r) | (x >> (32 - r));
}

// Threefry-2x32, 20 rounds, bit-exact with jax._src.prng.threefry2x32.
__host__ __device__ constexpr U2 threefry2x32(unsigned k0, unsigned k1,
                                              unsigned x0, unsigned x1) {
  const unsigned k2 = k0 ^ k1 ^ 0x1BD11BDAu;
  x0 += k0; x1 += k1;
  // rotations {13,15,26,6}
  x0 += x1; x1 = rotl32(x1, 13); x1 ^= x0;
  x0 += x1; x1 = rotl32(x1, 15); x1 ^= x0;
  x0 += x1; x1 = rotl32(x1, 26); x1 ^= x0;
  x0 += x1; x1 = rotl32(x1, 6);  x1 ^= x0;
  x0 += k1; x1 += k2 + 1u;
  // {17,29,16,24}
  x0 += x1; x1 = rotl32(x1, 17); x1 ^= x0;
  x0 += x1; x1 = rotl32(x1, 29); x1 ^= x0;
  x0 += x1; x1 = rotl32(x1, 16); x1 ^= x0;
  x0 += x1; x1 = rotl32(x1, 24); x1 ^= x0;
  x0 += k2; x1 += k0 + 2u;
  // {13,15,26,6}
  x0 += x1; x1 = rotl32(x1, 13); x1 ^= x0;
  x0 += x1; x1 = rotl32(x1, 15); x1 ^= x0;
  x0 += x1; x1 = rotl32(x1, 26); x1 ^= x0;
  x0 += x1; x1 = rotl32(x1, 6);  x1 ^= x0;
  x0 += k0; x1 += k1 + 3u;
  // {17,29,16,24}
  x0 += x1; x1 = rotl32(x1, 17); x1 ^= x0;
  x0 += x1; x1 = rotl32(x1, 29); x1 ^= x0;
  x0 += x1; x1 = rotl32(x1, 16); x1 ^= x0;
  x0 += x1; x1 = rotl32(x1, 24); x1 ^= x0;
  x0 += k1; x1 += k2 + 4u;
  // {13,15,26,6}
  x0 += x1; x1 = rotl32(x1, 13); x1 ^= x0;
  x0 += x1; x1 = rotl32(x1, 15); x1 ^= x0;
  x0 += x1; x1 = rotl32(x1, 26); x1 ^= x0;
  x0 += x1; x1 = rotl32(x1, 6);  x1 ^= x0;
  x0 += k2; x1 += k0 + 5u;
  return U2{x0, x1};
}

// jax.random.key(42) -> (0,42); split() = threefry over iota(4) paired (0,2),(1,3).
// keys = reshape(concat(x0_out, x1_out), (2,2)):
//   k_s = (pair0.a, pair1.a), k_n = (pair0.b, pair1.b). Computed at compile time.
constexpr U2 kPair0 = threefry2x32(0u, 42u, 0u, 2u);
constexpr U2 kPair1 = threefry2x32(0u, 42u, 1u, 3u);
constexpr unsigned KS0 = kPair0.a, KS1 = kPair1.a;   // scale key
constexpr unsigned KN0 = kPair0.b, KN1 = kPair1.b;   // normal-noise key

__device__ __forceinline__ float bits_to_u01(unsigned w) {
  // JAX _uniform: bitcast((bits>>9) | 0x3f800000) - 1.0  in [0,1)
  return __uint_as_float((w >> 9) | 0x3F800000u) - 1.0f;
}

// XLA ErfInv32 polynomial (what lax.erf_inv lowers to for f32 on GPU).
__device__ __forceinline__ float erfinv_f32(float x) {
  float w = -log1pf(-x * x);
  float p;
  if (w < 5.0f) {
    w = w - 2.5f;
    p = 2.81022636e-08f;
    p = fmaf(p, w, 3.43273939e-07f);
    p = fmaf(p, w, -3.5233877e-06f);
    p = fmaf(p, w, -4.39150654e-06f);
    p = fmaf(p, w, 0.00021858087f);
    p = fmaf(p, w, -0.00125372503f);
    p = fmaf(p, w, -0.00417768164f);
    p = fmaf(p, w, 0.246640727f);
    p = fmaf(p, w, 1.50140941f);
  } else {
    w = sqrtf(w) - 3.0f;
    p = -0.000200214257f;
    p = fmaf(p, w, 0.000100950558f);
    p = fmaf(p, w, 0.00134934322f);
    p = fmaf(p, w, -0.00367342844f);
    p = fmaf(p, w, 0.00573950773f);
    p = fmaf(p, w, -0.0076224613f);
    p = fmaf(p, w, 0.00943887047f);
    p = fmaf(p, w, 1.00167406f);
    p = fmaf(p, w, 2.83297682f);
  }
  return p * x;
}

// jax.random.normal: u = uniform(lo=nextafter(-1,0)=-0.99999994, hi=1); sqrt(2)*erfinv(u)
__device__ __forceinline__ float normal_from_bits(unsigned bits) {
  const float lo = -0.99999994f;           // 0xBF7FFFFF
  float u = bits_to_u01(bits);
  u = u * 2.0f + lo;                       // (hi - lo) rounds to 2.0f in f32
  u = fmaxf(u, lo);
  return 1.41421356237309504880f * erfinv_f32(u);
}

constexpr unsigned kB = 8192u;
constexpr unsigned kD = 4096u;
constexpr unsigned kN = kB * kD;           // 33554432
constexpr unsigned kHalf = kN / 2u;        // 16777216 (threefry counter pairing span)
constexpr unsigned kBlock = 256u;          // 8 wave32s per workgroup
constexpr unsigned kThreads = kHalf / 4u;  // each thread: float4 from each half

__global__ __launch_bounds__(kBlock) void gaussian_dropout_kernel(
    const float* __restrict__ x, float* __restrict__ out) {
  const unsigned t  = blockIdx.x * kBlock + threadIdx.x;
  const unsigned i0 = t * 4u;              // flat index in lower half (row-major)

  // ---- per-row scales: row index is wave-uniform -> scalarize (SALU threefry,
  // runs concurrently with the vector noise chain). Rows r0 and r0+4096 are a
  // single threefry counter pair in JAX's s ~ U[0,0.5) array of size 8192.
  const unsigned r0 = (unsigned)__builtin_amdgcn_readfirstlane((int)(i0 >> 12));
  const U2 sw = threefry2x32(KS0, KS1, r0, r0 + 4096u);
  const float s_lo = bits_to_u01(sw.a) * 0.5f;
  const float s_hi = bits_to_u01(sw.b) * 0.5f;

  // ---- CDNA5 async global->LDS staging of both x float4s (ASYNCcnt-tracked,
  // no VGPR destinations in flight while we compute the RNG below).
  __shared__ float4 tile[2u * kBlock];
  const unsigned lds_a = (unsigned)(uintptr_t)&tile[threadIdx.x];
  const unsigned lds_b = (unsigned)(uintptr_t)&tile[threadIdx.x + kBlock];
  const unsigned off_a = i0 * 4u;                       // byte offset, lower half
  const unsigned off_b = off_a + kHalf * 4u;            // byte offset, upper half
  const unsigned long long xbase = (unsigned long long)(uintptr_t)x;
  asm volatile(
      "global_load_async_to_lds_b128 %0, %1, %4\n\t"
      "global_load_async_to_lds_b128 %2, %3, %4"
      :
      : "v"(lds_a), "v"(off_a), "v"(lds_b), "v"(off_b), "s"(xbase)
      : "memory");

  // ---- Threefry noise for 8 elements (4 calls; each call's x0/x1 feed the
  // lower/upper half elements respectively, matching JAX's counter pairing).
  // This ~500-op VALU chain fully overlaps the async loads above.
  float na[4], nb[4];
#pragma unroll
  for (unsigned e = 0; e < 4u; ++e) {
    const unsigned idx = i0 + e;
    const U2 w = threefry2x32(KN0, KN1, idx, idx + kHalf);
    na[e] = normal_from_bits(w.a);
    nb[e] = normal_from_bits(w.b);
  }

  asm volatile("s_wait_asynccnt 0x0" ::: "memory");
  const float4 xa = tile[threadIdx.x];
  const float4 xb = tile[threadIdx.x + kBlock];

  float4 oa, ob;
  oa.x = xa.x * (na[0] * s_lo + 1.0f);
  oa.y = xa.y * (na[1] * s_lo + 1.0f);
  oa.z = xa.z * (na[2] * s_lo + 1.0f);
  oa.w = xa.w * (na[3] * s_lo + 1.0f);
  ob.x = xb.x * (nb[0] * s_hi + 1.0f);
  ob.y = xb.y * (nb[1] * s_hi + 1.0f);
  ob.z = xb.z * (nb[2] * s_hi + 1.0f);
  ob.w = xb.w * (nb[3] * s_hi + 1.0f);

  float4* o4 = reinterpret_cast<float4*>(out);
  o4[t] = oa;                 // global_store_b128
  o4[t + kHalf / 4u] = ob;    // global_store_b128
}

}  // namespace

extern "C" void kernel_launch(void* const* d_in, const int* in_sizes, int n_in,
                              void* d_out, int out_size, void* d_ws, size_t ws_size,
                              hipStream_t stream) {
  (void)in_sizes; (void)n_in; (void)d_ws; (void)ws_size; (void)out_size;
  const float* x = (const float*)d_in[0];
  float* out = (float*)d_out;
  const unsigned blocks = kThreads / kBlock;  // 16384
  gaussian_dropout_kernel<<<dim3(blocks), dim3(kBlock), 0, stream>>>(x, out);
}